// GNN_3874060501837
// MI455X (gfx1250) — compile-verified
//
#include <hip/hip_runtime.h>

typedef __attribute__((ext_vector_type(2))) float v2f;
typedef __attribute__((ext_vector_type(8))) float v8f;

#define BN_EPS 1e-5f

// ---------------- utility kernels ----------------

__global__ void zero_f32(float* __restrict__ p, int n) {
  int i = blockIdx.x * blockDim.x + threadIdx.x;
  if (i < n) p[i] = 0.0f;
}

__global__ void deg_count(const int* __restrict__ dst, float* __restrict__ deg, int E) {
  int e = blockIdx.x * blockDim.x + threadIdx.x;
  if (e < E) atomicAdd(&deg[dst[e]], 1.0f);
}

// deg -> D^{-1/2} including the self loop (+1)
__global__ void make_dinv(float* __restrict__ deg_dinv, int n) {
  int i = blockIdx.x * blockDim.x + threadIdx.x;
  if (i < n) deg_dinv[i] = rsqrtf(deg_dinv[i] + 1.0f);
}

__global__ void make_norm(const int* __restrict__ src, const int* __restrict__ dst,
                          const float* __restrict__ dinv, float* __restrict__ norm, int E) {
  int e = blockIdx.x * blockDim.x + threadIdx.x;
  if (e < E) norm[e] = dinv[src[e]] * dinv[dst[e]];
}

// zero-pad W4 (16x40) to 16x48 so the WMMA tiling stays 16-aligned
__global__ void pad_w4(const float* __restrict__ W4, float* __restrict__ wpad) {
  int i = blockIdx.x * blockDim.x + threadIdx.x;
  if (i < 16 * 48) {
    int k = i / 48, f = i - k * 48;
    wpad[i] = (f < 40) ? W4[k * 40 + f] : 0.0f;
  }
}

// ---------------- dense GEMM via V_WMMA_F32_16X16X4_F32 ----------------
// One wave (32 threads) computes a 16x16 tile of H = X(K-major) @ W(KxFOUT).
template <int K, int FOUT>
__global__ __launch_bounds__(32)
void gemm_wmma(const float* __restrict__ X, const float* __restrict__ W,
               float* __restrict__ H, int n) {
  const int lane = threadIdx.x;            // 0..31
  const int rb   = blockIdx.x;             // 16-row block
  const int cb   = blockIdx.y;             // 16-col block
  const int m    = lane & 15;
  const int half = lane >> 4;              // 0: K={0,1}, 1: K={2,3}
  const int kh   = half * 2;

  int row = rb * 16 + m;
  int lrow = (row < n) ? row : (n - 1);    // clamp reads; store is guarded
  const int col = cb * 16 + m;

  v8f c = {};
#pragma unroll
  for (int k0 = 0; k0 < K; k0 += 4) {
    v2f a, b;
    a.x = X[(long)lrow * K + (k0 + kh)];
    a.y = X[(long)lrow * K + (k0 + kh + 1)];
    b.x = W[(k0 + kh) * FOUT + col];
    b.y = W[(k0 + kh + 1) * FOUT + col];
    // 8 args: (neg_a, A, neg_b, B, c_mod, C, reuse_a, reuse_b)
    c = __builtin_amdgcn_wmma_f32_16x16x4_f32(false, a, false, b, (short)0, c,
                                              false, false);
  }

#pragma unroll
  for (int r = 0; r < 8; ++r) {
    int orow = rb * 16 + r + 8 * half;     // C/D layout: VGPR r -> M = r + 8*half
    if (orow < n) H[(long)orow * FOUT + col] = c[r];
  }
}

// ---------------- edge aggregation: agg[dst] += h[src] * norm ----------------
// One wave per edge (plus N self loops); lanes stride the feature dimension.
template <int FH, int FOUT>
__global__ void gcn_aggregate(const float* __restrict__ h, const int* __restrict__ src,
                              const int* __restrict__ dst, const float* __restrict__ norm,
                              const float* __restrict__ dinv, float* __restrict__ agg,
                              int E, int N) {
  int wid  = (int)((blockIdx.x * (long)blockDim.x + threadIdx.x) >> 5);
  int lane = threadIdx.x & 31;
  if (wid >= E + N) return;
  int s, d;
  float w;
  if (wid < E) {
    s = src[wid];
    d = dst[wid];
    w = norm[wid];
  } else {               // self loop
    s = wid - E;
    d = s;
    float di = dinv[s];
    w = di * di;
  }
  const float* hs = h + (long)s * FH;
  float* ad = agg + (long)d * FOUT;
#pragma unroll
  for (int f = lane; f < FOUT; f += 32) atomicAdd(&ad[f], hs[f] * w);
}

// ---------------- fused bias + BatchNorm(eval) + ReLU ----------------
__global__ void bias_bn_relu(const float* __restrict__ agg, const float* __restrict__ b,
                             const float* __restrict__ gamma, const float* __restrict__ beta,
                             const float* __restrict__ rmean, const float* __restrict__ rvar,
                             float* __restrict__ out, int total, int F) {
  int i = blockIdx.x * blockDim.x + threadIdx.x;
  if (i >= total) return;
  int f = i % F;
  float v = agg[i] + b[f];
  v = (v - rmean[f]) * rsqrtf(rvar[f] + BN_EPS) * gamma[f] + beta[f];
  out[i] = fmaxf(v, 0.0f);
}

// ---------------- fused bias + log_softmax over 40 classes ----------------
__global__ void bias_log_softmax40(const float* __restrict__ agg, const float* __restrict__ b,
                                   float* __restrict__ out, int n) {
  int i = blockIdx.x * blockDim.x + threadIdx.x;
  if (i >= n) return;
  float v[40];
  float mx = -3.402823466e38f;
#pragma unroll
  for (int f = 0; f < 40; ++f) {
    v[f] = agg[(long)i * 40 + f] + b[f];
    mx = fmaxf(mx, v[f]);
  }
  float s = 0.0f;
#pragma unroll
  for (int f = 0; f < 40; ++f) s += __expf(v[f] - mx);
  float ls = __logf(s);
#pragma unroll
  for (int f = 0; f < 40; ++f) out[(long)i * 40 + f] = v[f] - mx - ls;
}

// ---------------- driver ----------------

static inline int cdiv(long a, long b) { return (int)((a + b - 1) / b); }

extern "C" void kernel_launch(void* const* d_in, const int* in_sizes, int n_in,
                              void* d_out, int out_size, void* d_ws, size_t ws_size,
                              hipStream_t stream) {
  const float* x  = (const float*)d_in[0];
  const int*   ei = (const int*)d_in[1];
  const float* W1 = (const float*)d_in[2];  const float* b1 = (const float*)d_in[3];
  const float* W2 = (const float*)d_in[4];  const float* b2 = (const float*)d_in[5];
  const float* W3 = (const float*)d_in[6];  const float* b3 = (const float*)d_in[7];
  const float* W4 = (const float*)d_in[8];  const float* b4 = (const float*)d_in[9];
  const float* g1 = (const float*)d_in[10]; const float* be1 = (const float*)d_in[11];
  const float* rm1 = (const float*)d_in[12]; const float* rv1 = (const float*)d_in[13];
  const float* g2 = (const float*)d_in[14]; const float* be2 = (const float*)d_in[15];
  const float* rm2 = (const float*)d_in[16]; const float* rv2 = (const float*)d_in[17];
  const float* g3 = (const float*)d_in[18]; const float* be3 = (const float*)d_in[19];
  const float* rm3 = (const float*)d_in[20]; const float* rv3 = (const float*)d_in[21];

  const int N = in_sizes[0] / 64;   // 50000
  const int E = in_sizes[1] / 2;    // 800000
  const int* src = ei;
  const int* dst = ei + E;

  // workspace layout (floats)
  float* ws    = (float*)d_ws;
  float* dinv  = ws;                         // N
  float* norm  = dinv + N;                   // E
  float* wpad  = norm + E;                   // 16*48
  float* bufA  = wpad + 16 * 48;             // N*64
  float* bufB  = bufA + (long)N * 64;        // N*64

  const int T = 256;
  const int rowBlocks = (N + 15) / 16;       // 3125
  const long aggThreads = (long)(E + N) * 32;
  const int aggBlocks = cdiv(aggThreads, T);

  // --- normalization: deg -> dinv -> norm ---
  zero_f32<<<cdiv(N, T), T, 0, stream>>>(dinv, N);
  deg_count<<<cdiv(E, T), T, 0, stream>>>(dst, dinv, E);
  make_dinv<<<cdiv(N, T), T, 0, stream>>>(dinv, N);
  make_norm<<<cdiv(E, T), T, 0, stream>>>(src, dst, dinv, norm, E);
  pad_w4<<<cdiv(16 * 48, T), T, 0, stream>>>(W4, wpad);

  // --- layer 1: 64 -> 64 ---
  gemm_wmma<64, 64><<<dim3(rowBlocks, 4), 32, 0, stream>>>(x, W1, bufA, N);
  zero_f32<<<cdiv((long)N * 64, T), T, 0, stream>>>(bufB, N * 64);
  gcn_aggregate<64, 64><<<aggBlocks, T, 0, stream>>>(bufA, src, dst, norm, dinv, bufB, E, N);
  bias_bn_relu<<<cdiv((long)N * 64, T), T, 0, stream>>>(bufB, b1, g1, be1, rm1, rv1, bufA, N * 64, 64);

  // --- layer 2: 64 -> 32 ---
  gemm_wmma<64, 32><<<dim3(rowBlocks, 2), 32, 0, stream>>>(bufA, W2, bufB, N);
  zero_f32<<<cdiv((long)N * 32, T), T, 0, stream>>>(bufA, N * 32);
  gcn_aggregate<32, 32><<<aggBlocks, T, 0, stream>>>(bufB, src, dst, norm, dinv, bufA, E, N);
  bias_bn_relu<<<cdiv((long)N * 32, T), T, 0, stream>>>(bufA, b2, g2, be2, rm2, rv2, bufB, N * 32, 32);

  // --- layer 3: 32 -> 16 ---
  gemm_wmma<32, 16><<<dim3(rowBlocks, 1), 32, 0, stream>>>(bufB, W3, bufA, N);
  zero_f32<<<cdiv((long)N * 16, T), T, 0, stream>>>(bufB, N * 16);
  gcn_aggregate<16, 16><<<aggBlocks, T, 0, stream>>>(bufA, src, dst, norm, dinv, bufB, E, N);
  bias_bn_relu<<<cdiv((long)N * 16, T), T, 0, stream>>>(bufB, b3, g3, be3, rm3, rv3, bufA, N * 16, 16);

  // --- layer 4: 16 -> 40 (padded to 48 for the GEMM) ---
  gemm_wmma<16, 48><<<dim3(rowBlocks, 3), 32, 0, stream>>>(bufA, wpad, bufB, N);
  zero_f32<<<cdiv((long)N * 40, T), T, 0, stream>>>(bufA, N * 40);
  gcn_aggregate<48, 40><<<aggBlocks, T, 0, stream>>>(bufB, src, dst, norm, dinv, bufA, E, N);
  bias_log_softmax40<<<cdiv(N, T), T, 0, stream>>>(bufA, b4, (float*)d_out, N);
}